// PhaseCoherenceLoss_17643725652264
// MI455X (gfx1250) — compile-verified
//
#include <hip/hip_runtime.h>
#include <hip/hip_bf16.h>

// CDNA5 / gfx1250. Wave32.
typedef float v2f __attribute__((ext_vector_type(2)));
typedef float v8f __attribute__((ext_vector_type(8)));

#define T_DIM       8192
#define C_DIM       3
#define TILE_TS     256          // timesteps staged per block
#define TRANS_PB    (TILE_TS-1)  // transitions produced per block (overlap 1)
#define ILLEGAL_MASK 0xCCu       // bits 2,3,6,7 : (0,2),(1,0),(2,0),(2,1)
#define NSLOTS      256
#define WEIGHT_F    10.0f

// ---------------------------------------------------------------- zero ws
__global__ void pcl_zero_ws(float* __restrict__ wl, unsigned* __restrict__ wc) {
    int i = threadIdx.x;
    wl[i] = 0.0f;
    wc[i] = 0u;
}

// ------------------------------------------------- main streaming kernel
// One block = one row-chunk of 256 timesteps -> 255 transitions.
__global__ __launch_bounds__(TILE_TS)
void pcl_scan(const float* __restrict__ logits,
              float* __restrict__ wl, unsigned* __restrict__ wc,
              int chunks_per_row, int T) {
    __shared__ float    stile[TILE_TS * C_DIM];  // staged logits
    __shared__ float    smax[TILE_TS];           // per-timestep max logit
    __shared__ int      sidx[TILE_TS];           // per-timestep argmax
    __shared__ float    rl[8];
    __shared__ unsigned rc[8];

    const int tid   = threadIdx.x;
    const int row   = blockIdx.x / chunks_per_row;
    const int chunk = blockIdx.x % chunks_per_row;
    const int t0    = chunk * TRANS_PB;              // first timestep of tile

    // ---- stage tile: fully coalesced, non-temporal (single-use stream)
    const size_t gbase  = ((size_t)row * T + t0) * C_DIM;
    const int    nts    = min(TILE_TS, T - t0);      // valid timesteps in tile
    const int    nflt   = nts * C_DIM;
    #pragma unroll
    for (int i = tid; i < TILE_TS * C_DIM; i += TILE_TS) {
        stile[i] = (i < nflt) ? __builtin_nontemporal_load(&logits[gbase + i]) : 0.0f;
    }
    __syncthreads();

    // ---- per-timestep argmax (first-max tie-break, matches jnp.argmax)
    {
        float l0 = stile[tid * 3 + 0];
        float l1 = stile[tid * 3 + 1];
        float l2 = stile[tid * 3 + 2];
        float best = l0; int bi = 0;
        if (l1 > best) { best = l1; bi = 1; }
        if (l2 > best) { best = l2; bi = 2; }
        smax[tid] = best;
        sidx[tid] = bi;
    }
    __syncthreads();

    // ---- transitions t -> t+1  (t = t0+tid, need t <= T-2)
    float    lloss = 0.0f;
    unsigned lcnt  = 0u;
    if (tid < TRANS_PB && (t0 + tid) < (T - 1)) {
        int pt  = sidx[tid];
        int pt1 = sidx[tid + 1];
        if ((ILLEGAL_MASK >> (pt * 3 + pt1)) & 1u) {
            float g = smax[tid + 1];   // gathered = max logit at t+1
            lloss = g * g;
            lcnt  = 1u;
        }
    }

    // ---- wave32 reduce
    #pragma unroll
    for (int m = 16; m >= 1; m >>= 1) {
        lloss += __shfl_xor(lloss, m, 32);
        lcnt  += __shfl_xor(lcnt,  m, 32);
    }
    const int wid = tid >> 5, lane = tid & 31;
    if (lane == 0) { rl[wid] = lloss; rc[wid] = lcnt; }
    __syncthreads();

    // ---- block reduce + hashed-slot atomics (contention / 256)
    if (tid == 0) {
        float    L = 0.0f; unsigned Cn = 0u;
        #pragma unroll
        for (int i = 0; i < 8; ++i) { L += rl[i]; Cn += rc[i]; }
        if (Cn) {
            int slot = blockIdx.x & (NSLOTS - 1);
            atomicAdd(&wl[slot], L);
            atomicAdd(&wc[slot], Cn);
        }
    }
}

// ------------------------------------------- finalize: WMMA 256 -> scalar
// One wave (32 lanes). D = 1s(16x4) x B(4x16) + C : every row of D holds the
// column sums of B, so sum(lanes 0..15 of D vgpr0) == sum of all 64 B values.
__global__ __launch_bounds__(32)
void pcl_finalize(const float* __restrict__ wl, const unsigned* __restrict__ wc,
                  float* __restrict__ out) {
    const int ln = threadIdx.x;  // 0..31, EXEC all ones (required for WMMA)
    v2f ones; ones.x = 1.0f; ones.y = 1.0f;
    v8f accL = {}; v8f accC = {};
    #pragma unroll
    for (int c = 0; c < 4; ++c) {
        int base = c * 64 + ln * 2;
        v2f bl, bc;
        bl.x = wl[base];            bl.y = wl[base + 1];
        bc.x = (float)wc[base];     bc.y = (float)wc[base + 1];
        accL = __builtin_amdgcn_wmma_f32_16x16x4_f32(false, ones, false, bl,
                                                     (short)0, accL, false, false);
        accC = __builtin_amdgcn_wmma_f32_16x16x4_f32(false, ones, false, bc,
                                                     (short)0, accC, false, false);
    }
    // row M=0 lives in lanes 0..15 of D vgpr0; reduce across the 16-lane group.
    float L  = accL[0];
    float Cn = accC[0];
    #pragma unroll
    for (int m = 1; m < 16; m <<= 1) {
        L  += __shfl_xor(L,  m, 32);
        Cn += __shfl_xor(Cn, m, 32);
    }
    if (ln == 0) {
        out[0] = (Cn > 0.0f) ? (WEIGHT_F * L / Cn) : 0.0f;
    }
}

// ----------------------------------------------------------------- launch
extern "C" void kernel_launch(void* const* d_in, const int* in_sizes, int n_in,
                              void* d_out, int out_size, void* d_ws, size_t ws_size,
                              hipStream_t stream) {
    (void)n_in; (void)out_size; (void)ws_size;
    const float* logits = (const float*)d_in[0];
    float*    out = (float*)d_out;
    float*    wl  = (float*)d_ws;                 // 256 floats
    unsigned* wc  = ((unsigned*)d_ws) + NSLOTS;   // 256 uints

    const int T = T_DIM;
    const int B = in_sizes[0] / (T * C_DIM);      // 2048 for the reference
    const int chunks_per_row = (T - 1 + TRANS_PB - 1) / TRANS_PB;  // 33
    const int nblocks = B * chunks_per_row;

    pcl_zero_ws<<<1, NSLOTS, 0, stream>>>(wl, wc);
    pcl_scan<<<nblocks, TILE_TS, 0, stream>>>(logits, wl, wc, chunks_per_row, T);
    pcl_finalize<<<1, 32, 0, stream>>>(wl, wc, out);
}